// BatchIndependentLoss_69020124447275
// MI455X (gfx1250) — compile-verified
//
#include <hip/hip_runtime.h>
#include <hip/hip_bf16.h>
#include <math.h>

typedef __attribute__((ext_vector_type(2))) float v2f;
typedef __attribute__((ext_vector_type(8))) float v8f;

#define NTOT   4096          // B * V rows of the stacked contrast matrix
#define BROWS  2048          // B
#define DIMF   256           // feature dim
#define KSTEPS 64            // 256 / 4 (K per v_wmma_f32_16x16x4_f32)
#define INV_T  14.2857142857142857f   // 1 / 0.07
#define GAMMAF 0.9f

// contrast row i -> float offset into features[B, V, D] (views stacked: rows
// 0..B-1 = view 0, B..2B-1 = view 1)
__device__ __forceinline__ int frow_off(int i) {
    return ((i & (BROWS - 1)) * 2 + (i >> 11)) * DIMF;
}

// ---------------------------------------------------------------------------
// Kernel A: fused fp32 WMMA GEMM (S = F F^T / T) + online row-softmax stats.
// Wave = 16 rows; workgroup = 8 waves = 128 rows; gridDim.y = column splits.
// Two independent WMMA accumulators (even/odd K) break the 64-deep RAW chain
// on the accumulator so the matrix pipe stays fed at low wave occupancy.
// Emits per (row, split): running max m, neg exp-sum s, neg exp*logit sum w,
// raw positive logit p (pre max-subtraction).
// ---------------------------------------------------------------------------
__global__ __launch_bounds__(256)
void supcon_gemm_stats(const float* __restrict__ F,
                       float* __restrict__ ws_m, float* __restrict__ ws_s,
                       float* __restrict__ ws_w, float* __restrict__ ws_p,
                       int splits, int tilesPerSplit)
{
    const int lane = threadIdx.x & 31;
    const int wave = threadIdx.x >> 5;
    const int half = lane >> 4;      // 0: K lanes 0/1 + M 0..7 ; 1: K 2/3 + M 8..15
    const int nloc = lane & 15;

    const int rowbase = blockIdx.x * 128 + wave * 16;
    const int tile0   = blockIdx.y * tilesPerSplit;

    // A fragments: 16x256 panel resident in VGPRs for the whole kernel.
    // ISA layout (32-bit A 16x4): lanes 0-15 hold (M=lane, K=0..1),
    // lanes 16-31 hold (M=lane-16, K=2..3).
    const v2f* arow = (const v2f*)(F + frow_off(rowbase + nloc));
    v2f afrag[KSTEPS];
    #pragma unroll
    for (int k = 0; k < KSTEPS; ++k)
        afrag[k] = arow[2 * k + half];

    // online softmax state, per lane: rows M = r + 8*half, columns = this
    // lane's N-slot across all visited tiles
    float mS[8], sS[8], wS[8], pS[8];
    #pragma unroll
    for (int r = 0; r < 8; ++r) { mS[r] = -1e30f; sS[r] = 0.f; wS[r] = 0.f; pS[r] = 0.f; }

    // tile index holding this wave's positives: p(i) = (i + B) mod N, and the
    // 16 positives of a 16-aligned row block land on one tile's diagonal
    const int postile = ((rowbase + BROWS) & (NTOT - 1)) >> 4;

    for (int t = 0; t < tilesPerSplit; ++t) {
        const int ct = tile0 + t;
        const v2f* brow = (const v2f*)(F + frow_off(ct * 16 + nloc));
        if (t + 1 < tilesPerSplit) {
            // prefetch next B tile (global_prefetch_b8); F is L2-resident
            __builtin_prefetch((const void*)(F + frow_off((ct + 1) * 16 + nloc)), 0, 0);
        }

        v8f acc0 = {0.f, 0.f, 0.f, 0.f, 0.f, 0.f, 0.f, 0.f};
        v8f acc1 = {0.f, 0.f, 0.f, 0.f, 0.f, 0.f, 0.f, 0.f};
        #pragma unroll
        for (int kg = 0; kg < 8; ++kg) {
            v2f bfrag[8];
            #pragma unroll
            for (int kk = 0; kk < 8; ++kk)
                bfrag[kk] = brow[2 * (kg * 8 + kk) + half];
            #pragma unroll
            for (int kk = 0; kk < 8; kk += 2) {
                acc0 = __builtin_amdgcn_wmma_f32_16x16x4_f32(
                        false, afrag[kg * 8 + kk],     false, bfrag[kk],
                        (short)0, acc0, false, false);
                acc1 = __builtin_amdgcn_wmma_f32_16x16x4_f32(
                        false, afrag[kg * 8 + kk + 1], false, bfrag[kk + 1],
                        (short)0, acc1, false, false);
            }
        }
        v8f acc = acc0 + acc1;

        const bool tp = (ct == postile);
        #pragma unroll
        for (int r = 0; r < 8; ++r) {
            float x   = acc[r] * INV_T;                       // logit (pre max-sub)
            bool  isp = tp && (nloc == r + (half << 3));      // paired-view element
            float mo  = mS[r];
            float mn  = fmaxf(mo, x);
            float d   = mo - mn;                              // <= 0
            float sc  = __expf(d);
            float e   = isp ? 0.0f : __expf(x - mn);          // positives excluded
            wS[r] = sc * (wS[r] + d * sS[r]) + e * (x - mn);
            sS[r] = sc * sS[r] + e;
            mS[r] = mn;
            if (isp) pS[r] = x;
        }
    }

    // merge the 16 per-lane partial states of each half (xor masks stay
    // inside lanes 0-15 / 16-31)
    #pragma unroll
    for (int msk = 1; msk <= 8; msk <<= 1) {
        #pragma unroll
        for (int r = 0; r < 8; ++r) {
            float om = __shfl_xor(mS[r], msk, 32);
            float os = __shfl_xor(sS[r], msk, 32);
            float ow = __shfl_xor(wS[r], msk, 32);
            float op = __shfl_xor(pS[r], msk, 32);
            float mn = fmaxf(mS[r], om);
            float d1 = mS[r] - mn, d2 = om - mn;
            float e1 = __expf(d1), e2 = __expf(d2);
            wS[r] = e1 * (wS[r] + d1 * sS[r]) + e2 * (ow + d2 * os);
            sS[r] = e1 * sS[r] + e2 * os;
            mS[r] = mn;
            pS[r] += op;
        }
    }

    if (nloc == 0) {
        #pragma unroll
        for (int r = 0; r < 8; ++r) {
            int row = rowbase + (half << 3) + r;
            int idx = row * splits + blockIdx.y;
            ws_m[idx] = mS[r];
            ws_s[idx] = sS[r];
            ws_w[idx] = wS[r];
            ws_p[idx] = pS[r];
        }
    }
}

// ---------------------------------------------------------------------------
// Kernel B: merge splits per row, apply EMA buffer math and the cross-row
// u-denominator (row i uses s of row i mod B), reduce to the scalar loss.
// One workgroup of 1024 threads, 4 rows each.
// ---------------------------------------------------------------------------
__global__ __launch_bounds__(1024)
void supcon_finalize(const float* __restrict__ ws_m, const float* __restrict__ ws_s,
                     const float* __restrict__ ws_w, const float* __restrict__ ws_p,
                     const float* __restrict__ u, const int* __restrict__ index,
                     float* __restrict__ out, int splits)
{
    __shared__ float sh_s[NTOT];
    __shared__ float sh_red[1024];
    const int t = threadIdx.x;

    float mf[4], wf[4], pf[4];
    #pragma unroll
    for (int q = 0; q < 4; ++q) {
        int row = t + q * 1024;
        float m = -1e30f, s = 0.f, w = 0.f, p = 0.f;
        for (int k = 0; k < splits; ++k) {
            int   idx = row * splits + k;
            float mk = ws_m[idx], sk = ws_s[idx], wk = ws_w[idx], pk = ws_p[idx];
            float mn = fmaxf(m, mk);
            float d1 = m - mn, d2 = mk - mn;
            float e1 = __expf(d1), e2 = __expf(d2);
            w = e1 * (w + d1 * s) + e2 * (wk + d2 * sk);
            s = e1 * s + e2 * sk;
            m = mn;
            p += pk;
        }
        sh_s[row] = s;
        mf[q] = m; wf[q] = w; pf[q] = p;
    }
    __syncthreads();

    float accv = 0.f;
    #pragma unroll
    for (int q = 0; q < 4; ++q) {
        int row = t + q * 1024;
        int b   = row & (BROWS - 1);
        float un   = (1.0f - GAMMAF) * u[index[b]] + GAMMAF * sh_s[b];  // u_new[b]
        float posl = pf[q] - mf[q];                                     // logits[i, p(i)]
        accv += -(posl - wf[q] / un);                                   // T/BASE_T == 1
    }
    sh_red[t] = accv;
    __syncthreads();
    for (int off = 512; off > 0; off >>= 1) {
        if (t < off) sh_red[t] += sh_red[t + off];
        __syncthreads();
    }
    if (t == 0) out[0] = sh_red[0] * (1.0f / (float)NTOT);
}

// ---------------------------------------------------------------------------
extern "C" void kernel_launch(void* const* d_in, const int* in_sizes, int n_in,
                              void* d_out, int out_size, void* d_ws, size_t ws_size,
                              hipStream_t stream) {
    (void)in_sizes; (void)n_in; (void)out_size;
    const int*   index = (const int*)d_in[0];
    const float* feats = (const float*)d_in[1];
    const float* u     = (const float*)d_in[2];
    float* out = (float*)d_out;
    float* wsf = (float*)d_ws;

    // pick column splits that fit the workspace: 4 arrays x NTOT x splits floats
    int splits = 8;
    while (splits > 1 &&
           (size_t)4 * NTOT * (size_t)splits * sizeof(float) > ws_size)
        splits >>= 1;
    const int tilesPerSplit = (NTOT / 16) / splits;

    float* ws_m = wsf;
    float* ws_s = wsf + (size_t)NTOT * splits;
    float* ws_w = wsf + (size_t)2 * NTOT * splits;
    float* ws_p = wsf + (size_t)3 * NTOT * splits;

    dim3 grid(NTOT / 128, splits);
    supcon_gemm_stats<<<grid, 256, 0, stream>>>(feats, ws_m, ws_s, ws_w, ws_p,
                                                splits, tilesPerSplit);
    supcon_finalize<<<1, 1024, 0, stream>>>(ws_m, ws_s, ws_w, ws_p,
                                            u, index, out, splits);
}